// PhysicsLoss_20280835572490
// MI455X (gfx1250) — compile-verified
//
#include <hip/hip_runtime.h>

// Problem constants (match reference setup_inputs / reference())
#define BROWS 4096
#define TCOLS 6000
#define PICK_WIN 20
#define BETA 25.0f
#define W_PICK 0.1f

typedef __attribute__((ext_vector_type(2))) float v2f;
typedef __attribute__((ext_vector_type(8))) float v8f;

// ---------------------------------------------------------------------------
// Kernel 1: one block per row.
//  Phase 1: streaming float4 pass -> L1 partial sum + argmax(|target|) (first max).
//  Phase 2: wave 0 computes windowed soft-argmax for pred & target (<=40 elems),
//           using only intra-wave shuffles (all 32 lanes active).
// ---------------------------------------------------------------------------
__global__ __launch_bounds__(256) void physloss_row_kernel(
    const float* __restrict__ pred, const float* __restrict__ target,
    float* __restrict__ absOut, float* __restrict__ pickOut) {
  const int row = blockIdx.x;
  const int tid = threadIdx.x;
  const float* __restrict__ p = pred + (size_t)row * TCOLS;
  const float* __restrict__ t = target + (size_t)row * TCOLS;

  // ---- Phase 1: fused |p-t| sum and argmax(|t|) over the row ----
  float lsum = 0.0f;
  float bval = -1.0f;  // |t| >= 0, so -1 loses to everything
  int bidx = 0x7fffffff;

  const float4* __restrict__ p4 = (const float4*)p;  // rows are 16B aligned (T*4 = 24000)
  const float4* __restrict__ t4 = (const float4*)t;
  for (int v = tid; v < TCOLS / 4; v += 256) {
    float4 a = p4[v];
    float4 b = t4[v];
    lsum += fabsf(a.x - b.x) + fabsf(a.y - b.y) + fabsf(a.z - b.z) + fabsf(a.w - b.w);
    const int base = 4 * v;
    float m0 = fabsf(b.x), m1 = fabsf(b.y), m2 = fabsf(b.z), m3 = fabsf(b.w);
    // in-order strict '>' keeps first occurrence within the thread
    if (m0 > bval) { bval = m0; bidx = base + 0; }
    if (m1 > bval) { bval = m1; bidx = base + 1; }
    if (m2 > bval) { bval = m2; bidx = base + 2; }
    if (m3 > bval) { bval = m3; bidx = base + 3; }
  }

  __shared__ float sSum[256];
  __shared__ float sVal[256];
  __shared__ int   sIdx[256];
  sSum[tid] = lsum;
  sVal[tid] = bval;
  sIdx[tid] = bidx;
  __syncthreads();
  for (int s = 128; s > 0; s >>= 1) {
    if (tid < s) {
      sSum[tid] += sSum[tid + s];
      float v2 = sVal[tid + s];
      int i2 = sIdx[tid + s];
      // larger value wins; tie -> smaller index (matches jnp.argmax first-max)
      if (v2 > sVal[tid] || (v2 == sVal[tid] && i2 < sIdx[tid])) {
        sVal[tid] = v2;
        sIdx[tid] = i2;
      }
    }
    __syncthreads();
  }
  const int center = sIdx[0];
  const float rowAbs = sSum[0];

  // ---- Phase 2: windowed soft-argmax on [lo, hi), wave 0 only ----
  if (tid < 32) {
    int lo = center - PICK_WIN; if (lo < 0) lo = 0;
    int hi = center + PICK_WIN; if (hi > TCOLS) hi = TCOLS;

    const int j1 = lo + tid;
    const int j2 = lo + tid + 32;
    const bool ok1 = (j1 < hi);
    const bool ok2 = (j2 < hi);
    const float NEG = -__builtin_inff();

    float lp1 = ok1 ? BETA * fabsf(p[j1]) : NEG;
    float lt1 = ok1 ? BETA * fabsf(t[j1]) : NEG;
    float lp2 = ok2 ? BETA * fabsf(p[j2]) : NEG;
    float lt2 = ok2 ? BETA * fabsf(t[j2]) : NEG;

    // wave max (softmax stabilization, same as jax.nn.softmax)
    float mp = fmaxf(lp1, lp2);
    float mt = fmaxf(lt1, lt2);
    for (int o = 16; o > 0; o >>= 1) {
      mp = fmaxf(mp, __shfl_xor(mp, o, 32));
      mt = fmaxf(mt, __shfl_xor(mt, o, 32));
    }

    float ep1 = ok1 ? __expf(lp1 - mp) : 0.0f;
    float ep2 = ok2 ? __expf(lp2 - mp) : 0.0f;
    float et1 = ok1 ? __expf(lt1 - mt) : 0.0f;
    float et2 = ok2 ? __expf(lt2 - mt) : 0.0f;

    float sp  = ep1 + ep2;
    float spj = ep1 * (float)j1 + ep2 * (float)j2;
    float st  = et1 + et2;
    float stj = et1 * (float)j1 + et2 * (float)j2;
    for (int o = 16; o > 0; o >>= 1) {
      sp  += __shfl_xor(sp, o, 32);
      spj += __shfl_xor(spj, o, 32);
      st  += __shfl_xor(st, o, 32);
      stj += __shfl_xor(stj, o, 32);
    }

    if (tid == 0) {
      const float pickP = spj / sp;
      const float pickT = stj / st;
      absOut[row]  = rowAbs;
      pickOut[row] = fabsf(pickP - pickT);
    }
  }
}

// ---------------------------------------------------------------------------
// Kernel 2: single wave. Exact fp32 reduction of the 4096 per-row partials via
// V_WMMA_F32_16X16X4_F32:  C += A(16x4 data chunk) x B(4x16 ones).
// A layout (32-bit 16x4): lanes 0-15 hold M=lane with K=0 (V0), K=1 (V1);
// lanes 16-31 hold M=lane-16 with K=2 (V0), K=3 (V1). So value index
// idx = 4*M + K  ->  lane loads the float2 at 4*(lane&15) + 2*(lane>>4).
// D layout: VGPR r, lanes 0-15 = C[r, n]; lanes 16-31 = C[8+r, n].
// total = sum_m rowsum_m = (lane-local sum of 8 VGPRs) + shfl_xor(.,16).
// ---------------------------------------------------------------------------
__global__ __launch_bounds__(32) void physloss_finalize_kernel(
    const float* __restrict__ absIn, const float* __restrict__ pickIn,
    float* __restrict__ out) {
  const int lane = threadIdx.x;
  const int off = 4 * (lane & 15) + 2 * (lane >> 4);

  const v2f ones = {1.0f, 1.0f};
  v8f c1 = {};  // abs-loss partials
  v8f c2 = {};  // pick-loss partials

  for (int b = 0; b < BROWS; b += 64) {
    v2f a1, a2;
    const float* s1 = absIn + b + off;
    const float* s2 = pickIn + b + off;
    a1.x = s1[0]; a1.y = s1[1];
    a2.x = s2[0]; a2.y = s2[1];
    c1 = __builtin_amdgcn_wmma_f32_16x16x4_f32(false, a1, false, ones, (short)0, c1,
                                               false, false);
    c2 = __builtin_amdgcn_wmma_f32_16x16x4_f32(false, a2, false, ones, (short)0, c2,
                                               false, false);
  }

  float s1 = 0.0f, s2 = 0.0f;
  for (int r = 0; r < 8; ++r) { s1 += c1[r]; s2 += c2[r]; }
  s1 += __shfl_xor(s1, 16, 32);
  s2 += __shfl_xor(s2, 16, 32);

  if (lane == 0) {
    const float loss = s1 / ((float)BROWS * (float)TCOLS);
    const float pick_loss = (s2 / (float)BROWS) / (float)TCOLS;
    out[0] = loss + W_PICK * pick_loss;
  }
}

extern "C" void kernel_launch(void* const* d_in, const int* in_sizes, int n_in,
                              void* d_out, int out_size, void* d_ws, size_t ws_size,
                              hipStream_t stream) {
  (void)in_sizes; (void)n_in; (void)out_size; (void)ws_size;
  const float* pred = (const float*)d_in[0];
  const float* target = (const float*)d_in[1];
  float* out = (float*)d_out;

  float* absOut = (float*)d_ws;          // 4096 floats
  float* pickOut = absOut + BROWS;       // 4096 floats (32 KB total)

  physloss_row_kernel<<<BROWS, 256, 0, stream>>>(pred, target, absOut, pickOut);
  physloss_finalize_kernel<<<1, 32, 0, stream>>>(absOut, pickOut, out);
}